// TransformerConv_16836271800955
// MI455X (gfx1250) — compile-verified
//
#include <hip/hip_runtime.h>
#include <hip/hip_bf16.h>

typedef __attribute__((ext_vector_type(2))) float v2f;
typedef __attribute__((ext_vector_type(8))) float v8f;

#define INC 64   // input channels
#define HC  64   // H*C
#define NH  4    // heads
#define CCH 16   // channels per head
#define ED  32   // edge feature dim

__device__ __forceinline__ v8f wmma4(v2f a, v2f b, v8f c) {
  // V_WMMA_F32_16X16X4_F32: D = A(16x4) * B(4x16) + C(16x16)
  return __builtin_amdgcn_wmma_f32_16x16x4_f32(false, a, false, b, (short)0, c,
                                               false, false);
}

__device__ __forceinline__ void atomicMaxFloat(float* addr, float val) {
  // -inf-initialized float max via integer atomics (IEEE ordering trick)
  if (val >= 0.0f)
    atomicMax((int*)addr, __float_as_int(val));
  else
    atomicMin((unsigned int*)addr, __float_as_uint(val));
}

// ---------------------------------------------------------------------------
// Kernel 1: q/k/v = x @ W{q,k,v} + b   (WMMA f32 16x16x4, 16 K-steps, 4 N-tiles)
// ---------------------------------------------------------------------------
__global__ __launch_bounds__(256) void node_proj_kernel(
    const float* __restrict__ x,
    const float* Wq, const float* bq,
    const float* Wk, const float* bk,
    const float* Wv, const float* bv,
    float* q, float* k, float* v,
    int nNodes, int nTiles) {
  const int wave = threadIdx.x >> 5;
  const int lane = threadIdx.x & 31;
  const int tile = blockIdx.x * 8 + wave;
  if (tile >= nTiles) return;            // wave-uniform exit; EXEC stays all-1s
  const int row0  = tile * 16;
  const int mrow  = lane & 15;
  const int khalf = lane >> 4;           // lanes 16-31 carry K+2

  const int lrow = min(row0 + mrow, nNodes - 1);   // clamp loads on tail tile
  const float* xrow = x + (size_t)lrow * INC;

  v2f afrag[16];
#pragma unroll
  for (int ks = 0; ks < 16; ++ks) {
    const int kb = ks * 4 + khalf * 2;
    v2f a = { xrow[kb], xrow[kb + 1] };
    afrag[ks] = a;
  }

  const float* Wmat[3] = { Wq, Wk, Wv };
  const float* Bvec[3] = { bq, bk, bv };
  float*       Omat[3] = { q, k, v };

#pragma unroll
  for (int m = 0; m < 3; ++m) {
    const float* w = Wmat[m];
    v8f acc[4];
#pragma unroll
    for (int t = 0; t < 4; ++t) {
      const float bias = Bvec[m][t * 16 + mrow];
#pragma unroll
      for (int j = 0; j < 8; ++j) acc[t][j] = bias;
    }
#pragma unroll
    for (int ks = 0; ks < 16; ++ks) {
      const int kb = ks * 4 + khalf * 2;
#pragma unroll
      for (int t = 0; t < 4; ++t) {
        const int ncol = t * 16 + mrow;
        v2f b = { w[(size_t)kb * HC + ncol], w[(size_t)(kb + 1) * HC + ncol] };
        acc[t] = wmma4(afrag[ks], b, acc[t]);
      }
    }
    float* o = Omat[m];
#pragma unroll
    for (int t = 0; t < 4; ++t) {
#pragma unroll
      for (int j = 0; j < 8; ++j) {
        const int r = row0 + j + 8 * khalf;        // C/D layout: rows M / M+8
        if (r < nNodes) o[(size_t)r * HC + t * 16 + mrow] = acc[t][j];
      }
    }
  }
}

// ---------------------------------------------------------------------------
// Kernel 2: e = edge_attr @ We   (K=32 -> 8 WMMA K-steps, 4 N-tiles)
// ---------------------------------------------------------------------------
__global__ __launch_bounds__(256) void edge_proj_kernel(
    const float* __restrict__ ea, const float* __restrict__ We,
    float* __restrict__ e, int nEdges, int nTiles) {
  const int wave = threadIdx.x >> 5;
  const int lane = threadIdx.x & 31;
  const int tile = blockIdx.x * 8 + wave;
  if (tile >= nTiles) return;
  const int row0  = tile * 16;
  const int mrow  = lane & 15;
  const int khalf = lane >> 4;

  const int lrow = min(row0 + mrow, nEdges - 1);
  const float* arow = ea + (size_t)lrow * ED;

  v2f afrag[8];
#pragma unroll
  for (int ks = 0; ks < 8; ++ks) {
    const int kb = ks * 4 + khalf * 2;
    v2f a = { arow[kb], arow[kb + 1] };
    afrag[ks] = a;
  }

  v8f acc[4];
#pragma unroll
  for (int t = 0; t < 4; ++t)
#pragma unroll
    for (int j = 0; j < 8; ++j) acc[t][j] = 0.0f;

#pragma unroll
  for (int ks = 0; ks < 8; ++ks) {
    const int kb = ks * 4 + khalf * 2;
#pragma unroll
    for (int t = 0; t < 4; ++t) {
      const int ncol = t * 16 + mrow;
      v2f b = { We[(size_t)kb * HC + ncol], We[(size_t)(kb + 1) * HC + ncol] };
      acc[t] = wmma4(afrag[ks], b, acc[t]);
    }
  }
#pragma unroll
  for (int t = 0; t < 4; ++t) {
#pragma unroll
    for (int j = 0; j < 8; ++j) {
      const int r = row0 + j + 8 * khalf;
      if (r < nEdges) e[(size_t)r * HC + t * 16 + mrow] = acc[t][j];
    }
  }
}

// ---------------------------------------------------------------------------
__global__ void init_kernel(float* amax, float* denom, float* accum, int n) {
  const long long i = (long long)blockIdx.x * blockDim.x + threadIdx.x;
  if (i < (long long)n * NH) {
    amax[i]  = -__builtin_inff();
    denom[i] = 0.0f;
  }
  if (i < (long long)n * HC) accum[i] = 0.0f;
}

// ---------------------------------------------------------------------------
__global__ void alpha_kernel(const long long* __restrict__ ei,
                             const float* __restrict__ q,
                             const float* __restrict__ k,
                             const float* __restrict__ e,
                             float* __restrict__ alpha,
                             float* __restrict__ amax, long long nE) {
  const long long eh = (long long)blockIdx.x * blockDim.x + threadIdx.x;
  if (eh >= nE * NH) return;
  const long long edge = eh >> 2;
  const int h   = (int)(eh & 3);
  const int src = (int)ei[edge];
  const int dst = (int)ei[nE + edge];

  if (edge + 256 < nE)
    __builtin_prefetch(e + (size_t)(edge + 256) * HC + h * CCH, 0, 0);

  const float4* qr = (const float4*)(q + (size_t)dst * HC + h * CCH);
  const float4* kr = (const float4*)(k + (size_t)src * HC + h * CCH);
  const float4* er = (const float4*)(e + (size_t)edge * HC + h * CCH);
  float s = 0.0f;
#pragma unroll
  for (int c4 = 0; c4 < 4; ++c4) {
    const float4 qv = qr[c4], kv = kr[c4], ev = er[c4];
    s += qv.x * (kv.x + ev.x) + qv.y * (kv.y + ev.y) +
         qv.z * (kv.z + ev.z) + qv.w * (kv.w + ev.w);
  }
  s *= 0.25f;  // 1/sqrt(16)
  alpha[eh] = s;
  atomicMaxFloat(amax + (size_t)dst * NH + h, s);
}

// ---------------------------------------------------------------------------
__global__ void expsum_kernel(const long long* __restrict__ ei,
                              float* __restrict__ alpha,
                              const float* __restrict__ amax,
                              float* __restrict__ denom, long long nE) {
  const long long eh = (long long)blockIdx.x * blockDim.x + threadIdx.x;
  if (eh >= nE * NH) return;
  const long long edge = eh >> 2;
  const int h   = (int)(eh & 3);
  const int dst = (int)ei[nE + edge];
  const float ex = __expf(alpha[eh] - amax[(size_t)dst * NH + h]);
  alpha[eh] = ex;
  atomicAdd(denom + (size_t)dst * NH + h, ex);
}

// ---------------------------------------------------------------------------
__global__ void aggregate_kernel(const long long* __restrict__ ei,
                                 const float* __restrict__ v,
                                 const float* __restrict__ e,
                                 const float* __restrict__ alpha,
                                 const float* __restrict__ denom,
                                 float* __restrict__ accum, long long nE) {
  const long long eh = (long long)blockIdx.x * blockDim.x + threadIdx.x;
  if (eh >= nE * NH) return;
  const long long edge = eh >> 2;
  const int h   = (int)(eh & 3);
  const int src = (int)ei[edge];
  const int dst = (int)ei[nE + edge];
  const float w = alpha[eh] / (denom[(size_t)dst * NH + h] + 1e-16f);
  const float4* vr = (const float4*)(v + (size_t)src * HC + h * CCH);
  const float4* er = (const float4*)(e + (size_t)edge * HC + h * CCH);
  float* ar = accum + (size_t)dst * HC + h * CCH;
#pragma unroll
  for (int c4 = 0; c4 < 4; ++c4) {
    const float4 vv = vr[c4], ev = er[c4];
    atomicAdd(ar + 4 * c4 + 0, w * (vv.x + ev.x));
    atomicAdd(ar + 4 * c4 + 1, w * (vv.y + ev.y));
    atomicAdd(ar + 4 * c4 + 2, w * (vv.z + ev.z));
    atomicAdd(ar + 4 * c4 + 3, w * (vv.w + ev.w));
  }
}

// ---------------------------------------------------------------------------
__global__ void finalize_kernel(const float* __restrict__ x,
                                const float* __restrict__ Ws,
                                const float* __restrict__ bs,
                                const float* __restrict__ accum,
                                float* __restrict__ out, int n) {
  const int i = blockIdx.x * blockDim.x + threadIdx.x;
  if (i >= n * CCH) return;
  const int node = i / CCH;
  const int c    = i % CCH;
  const float* ar = accum + (size_t)node * HC + c;
  const float m = 0.25f * (ar[0] + ar[16] + ar[32] + ar[48]);
  const float* xr = x + (size_t)node * INC;
  float s = bs[c];
#pragma unroll 8
  for (int j = 0; j < INC; ++j) s += xr[j] * Ws[j * CCH + c];
  out[i] = m + s;
}

// ---------------------------------------------------------------------------
extern "C" void kernel_launch(void* const* d_in, const int* in_sizes, int n_in,
                              void* d_out, int out_size, void* d_ws,
                              size_t ws_size, hipStream_t stream) {
  const float*     x  = (const float*)d_in[0];
  const long long* ei = (const long long*)d_in[1];   // int64 edge_index [2,E]
  const float*     ea = (const float*)d_in[2];
  const float*     Wq = (const float*)d_in[3];
  const float*     bq = (const float*)d_in[4];
  const float*     Wk = (const float*)d_in[5];
  const float*     bk = (const float*)d_in[6];
  const float*     Wv = (const float*)d_in[7];
  const float*     bv = (const float*)d_in[8];
  const float*     We = (const float*)d_in[9];
  const float*     Ws = (const float*)d_in[10];
  const float*     bs = (const float*)d_in[11];
  float*           out = (float*)d_out;

  const int       N  = in_sizes[0] / INC;
  const long long E  = (long long)in_sizes[1] / 2;
  const int       Ei = (int)E;

  // Workspace layout (floats): q | k | v | e | alpha | amax | denom | accum
  float* ws    = (float*)d_ws;
  float* q     = ws;
  float* k     = q + (size_t)N * HC;
  float* v     = k + (size_t)N * HC;
  float* e     = v + (size_t)N * HC;
  float* alpha = e + (size_t)E * HC;
  float* amax  = alpha + (size_t)E * NH;
  float* denom = amax + (size_t)N * NH;
  float* accum = denom + (size_t)N * NH;

  const int nTilesN = (N + 15) / 16;
  const int nTilesE = (int)((E + 15) / 16);
  const long long ehTotal = E * NH;
  const int ehBlocks   = (int)((ehTotal + 255) / 256);
  const int initBlocks = (int)(((long long)N * HC + 255) / 256);

  node_proj_kernel<<<(nTilesN + 7) / 8, 256, 0, stream>>>(
      x, Wq, bq, Wk, bk, Wv, bv, q, k, v, N, nTilesN);
  edge_proj_kernel<<<(nTilesE + 7) / 8, 256, 0, stream>>>(ea, We, e, Ei,
                                                          nTilesE);
  init_kernel<<<initBlocks, 256, 0, stream>>>(amax, denom, accum, N);
  alpha_kernel<<<ehBlocks, 256, 0, stream>>>(ei, q, k, e, alpha, amax, E);
  expsum_kernel<<<ehBlocks, 256, 0, stream>>>(ei, alpha, amax, denom, E);
  aggregate_kernel<<<ehBlocks, 256, 0, stream>>>(ei, v, e, alpha, denom, accum,
                                                 E);
  finalize_kernel<<<(N * CCH + 255) / 256, 256, 0, stream>>>(x, Ws, bs, accum,
                                                             out, N);
}